// ChamferDistance_5738076307589
// MI455X (gfx1250) — compile-verified
//
#include <hip/hip_runtime.h>

// WMMA operand vector types (16x16x4 f32: A/B = 2 floats/lane, C/D = 8 floats/lane)
typedef float v2f __attribute__((ext_vector_type(2)));
typedef float v4f __attribute__((ext_vector_type(4)));
typedef float v8f __attribute__((ext_vector_type(8)));

#define WAVES   4                 // waves per block (wave32)
#define BLOCK   (WAVES * 32)
#define CHUNKP  4096              // staged target points per chunk -> 64 KB LDS

// One wave owns a 16-row strip of `query`; scans all of `target` in 16-col tiles.
// Distance tile computed entirely inside the WMMA accumulate:
//   C seed = ||q_row||^2 + ||t_col||^2,  A = -2*q (16x4),  B = t (4x16)
//   D[r][c] = ||q_r||^2 + ||t_c||^2 - 2 q_r . t_c  (exact squared distance)
__global__ __launch_bounds__(BLOCK)
void chamfer_nn_wmma(const float* __restrict__ query,
                     const float* __restrict__ target,
                     float* __restrict__ distOut,
                     float* __restrict__ idxOut,
                     int nq, int nt)
{
    __shared__ v4f s_pts[CHUNKP];   // {x, y, z, ||p||^2}

    const int  tid     = threadIdx.x;
    const int  lane    = tid & 31;
    const int  colLane = lane & 15;
    const bool hiHalf  = (lane >= 16);
    const int  wave    = tid >> 5;
    const int  b       = blockIdx.y;
    const int  rowBase = (blockIdx.x * WAVES + wave) * 16;

    const float* q = query  + (size_t)b * nq * 3;
    const float* t = target + (size_t)b * nt * 3;

    // Per-lane query point: lanes 0..15 <-> rows 0..15 (lanes 16..31 mirror)
    const int   qrow = rowBase + colLane;
    const float qx = q[qrow * 3 + 0];
    const float qy = q[qrow * 3 + 1];
    const float qz = q[qrow * 3 + 2];
    const float n1v = qx * qx + qy * qy + qz * qz;   // ||q||^2 (lanes 0..15 hold rows 0..15)

    // A-matrix (32-bit A 16x4): lanes 0-15 -> K=0,1 for M=lane;
    //                           lanes 16-31 -> K=2,3 for M=lane-16 (K=3 pad = 0)
    // Branchless build -> v_cndmask, keeps EXEC all-ones.
    v2f a;
    a.x = hiHalf ? (-2.0f * qz) : (-2.0f * qx);
    a.y = hiHalf ? 0.0f         : (-2.0f * qy);

    // Per-row ||q||^2 for the C/D layout: VGPR r holds row r (lanes 0-15)
    // or row r+8 (lanes 16-31). Fetch once via shuffles.
    float nr[8];
    #pragma unroll
    for (int r = 0; r < 8; ++r)
        nr[r] = __shfl(n1v, (hiHalf ? 8 : 0) + r, 32);

    float best[8];
    int   bidx[8];
    #pragma unroll
    for (int r = 0; r < 8; ++r) { best[r] = 3.4e38f; bidx[r] = 0; }

    for (int cb = 0; cb < nt; cb += CHUNKP) {
        __syncthreads();
        // Cooperative stage of target chunk (+ squared norms) into LDS
        for (int i = tid; i < CHUNKP; i += BLOCK) {
            const int p = cb + i;
            const float x = t[p * 3 + 0];
            const float y = t[p * 3 + 1];
            const float z = t[p * 3 + 2];
            v4f v; v.x = x; v.y = y; v.z = z; v.w = x * x + y * y + z * z;
            s_pts[i] = v;
        }
        __syncthreads();

        #pragma unroll 4
        for (int tt = 0; tt < CHUNKP / 16; ++tt) {
            // Single ds_load_b128; all components consumed unconditionally below.
            const v4f p = s_pts[tt * 16 + colLane];

            // B-matrix (4x16): lanes 0-15 -> K=0,1 for N=lane; 16-31 -> K=2,3
            v2f bb;
            bb.x = hiHalf ? p.z : p.x;
            bb.y = hiHalf ? 0.0f : p.y;

            v8f c;
            #pragma unroll
            for (int r = 0; r < 8; ++r) c[r] = p.w + nr[r];   // ||q_r||^2 + ||t_c||^2

            c = __builtin_amdgcn_wmma_f32_16x16x4_f32(false, a, false, bb,
                                                      (short)0, c, false, false);

            const int col = cb + tt * 16 + colLane;
            // Per-row running min/argmin (strict < keeps first occurrence per lane slot)
            #pragma unroll
            for (int r = 0; r < 8; ++r) {
                if (c[r] < best[r]) { best[r] = c[r]; bidx[r] = col; }
            }
        }
    }

    // Cross-lane min+argmin within each 16-lane group (rows 0-7 in lanes 0-15,
    // rows 8-15 in lanes 16-31); xor masks < 16 stay inside each group.
    #pragma unroll
    for (int m = 8; m >= 1; m >>= 1) {
        #pragma unroll
        for (int r = 0; r < 8; ++r) {
            const float od = __shfl_xor(best[r], m, 32);
            const int   oi = __shfl_xor(bidx[r], m, 32);
            if (od < best[r] || (od == best[r] && oi < bidx[r])) {
                best[r] = od; bidx[r] = oi;
            }
        }
    }

    // Writers: lane 0 -> rows 0..7, lane 16 -> rows 8..15
    #pragma unroll
    for (int r = 0; r < 8; ++r) {
        if (colLane == 0) {
            const int row = rowBase + r + (hiHalf ? 8 : 0);
            distOut[(size_t)b * nq + row] = fmaxf(best[r], 0.0f);  // guard tiny negatives
            idxOut [(size_t)b * nq + row] = (float)bidx[r];        // indices exact in fp32
        }
    }
}

extern "C" void kernel_launch(void* const* d_in, const int* in_sizes, int n_in,
                              void* d_out, int out_size, void* d_ws, size_t ws_size,
                              hipStream_t stream) {
    const float* xyz1 = (const float*)d_in[0];
    const float* xyz2 = (const float*)d_in[1];

    const int B = 4;
    const int N = in_sizes[0] / (B * 3);   // 8192
    const int M = in_sizes[1] / (B * 3);   // 8192

    float* out   = (float*)d_out;
    float* dist1 = out;                                        // [B,N]
    float* dist2 = out + (size_t)B * N;                        // [B,M]
    float* idx1  = out + (size_t)B * N + (size_t)B * M;        // [B,N] (as float)
    float* idx2  = idx1 + (size_t)B * N;                       // [B,M] (as float)

    dim3 grid1(N / (16 * WAVES), B);
    dim3 grid2(M / (16 * WAVES), B);

    // Direction 1: for each xyz1 point, nearest in xyz2
    chamfer_nn_wmma<<<grid1, BLOCK, 0, stream>>>(xyz1, xyz2, dist1, idx1, N, M);
    // Direction 2: for each xyz2 point, nearest in xyz1
    chamfer_nn_wmma<<<grid2, BLOCK, 0, stream>>>(xyz2, xyz1, dist2, idx2, M, N);
}